// Attention_86552180949092
// MI455X (gfx1250) — compile-verified
//
#include <hip/hip_runtime.h>
#include <math.h>

// ---------------------------------------------------------------------------
// Channel attention, algebraically factorized:
//   G_b   = X_b^T X_b                      [256,256]  (K = 4096 spatial)
//   S_bh  = scale * Wq_h^T G_b Wk_h        [64,64]
//   A_bh  = softmax_rows(S_bh)
//   P_bh  = A_bh^T W_out_h                 [64,256]
//   M_b   = sum_h Wv_h P_bh                [256,256]
//   y_b   = X_b M_b + b_out                [4096,256]
// All GEMMs run on V_WMMA_F32_16X16X4_F32 (full fp32; op is HBM-bound).
// K4 streams its K-panels with double-buffered GLOBAL_LOAD_ASYNC_TO_LDS_B128
// when the toolchain exposes the builtin (guarded; falls back to LDS stores).
// ---------------------------------------------------------------------------

typedef float v2f __attribute__((ext_vector_type(2)));
typedef float v8f __attribute__((ext_vector_type(8)));
typedef int   v4i __attribute__((vector_size(4 * sizeof(int))));

#define B_    16
#define HW_   4096
#define C_    256
#define HID_  512
#define NH_   8
#define DH_   64
#define W3_   1536          // 3*HID
#define SCALE_ 0.125f       // 64^-0.5

#if __has_builtin(__builtin_amdgcn_global_load_async_to_lds_b128) && \
    __has_builtin(__builtin_amdgcn_s_wait_asynccnt)
#define HAVE_ASYNC 1
#else
#define HAVE_ASYNC 0
#endif

static __device__ __forceinline__ v8f wmma4(v2f a, v2f b, v8f c) {
  // D = A(16x4,f32) * B(4x16,f32) + C(16x16,f32)
  return __builtin_amdgcn_wmma_f32_16x16x4_f32(false, a, false, b, (short)0, c,
                                               false, false);
}

// 16B global -> LDS copy: async (ASYNCcnt) when available, else direct store.
static __device__ __forceinline__ void cp16(const float* g, float* l) {
#if HAVE_ASYNC
  __builtin_amdgcn_global_load_async_to_lds_b128(
      (__attribute__((address_space(1))) v4i*)g,
      (__attribute__((address_space(3))) v4i*)l, 0, 0);
#else
  *(float4*)l = *(const float4*)g;
#endif
}

static __device__ __forceinline__ void wait_cp() {
#if HAVE_ASYNC
  __builtin_amdgcn_s_wait_asynccnt(0);
#endif
}

// ---------------------------------------------------------------------------
// K1: G_b = X_b^T X_b.  Block = 128 thr (4 waves) -> 64x64 tile of G.
// Tiles staged TRANSPOSED ([col][k], pitch 34) so every WMMA fragment is one
// aligned contiguous b64 LDS load (no operand-pack movs).
// ---------------------------------------------------------------------------
__global__ __launch_bounds__(128) void gram_kernel(const float* __restrict__ x,
                                                   float* __restrict__ G) {
  __shared__ float XiT[64][34];   // [channel][d]
  __shared__ float XjT[64][34];
  const int b = blockIdx.z, ti = blockIdx.y, tj = blockIdx.x;
  const int t = threadIdx.x, lane = t & 31, w = t >> 5;
  const int m = lane & 15, hl = lane >> 4, kb = hl * 2;
  const float* xb = x + (size_t)b * HW_ * C_;

  v8f acc[4] = {};
  for (int d0 = 0; d0 < HW_; d0 += 32) {
    __syncthreads();
    #pragma unroll
    for (int s = 0; s < 16; ++s) {           // stage 32 d-rows x 64 channels
      int idx = t + s * 128;
      int r = idx >> 6, c = idx & 63;        // r = d, c = channel
      XiT[c][r] = xb[(size_t)(d0 + r) * C_ + ti * 64 + c];
      XjT[c][r] = xb[(size_t)(d0 + r) * C_ + tj * 64 + c];
    }
    __syncthreads();
    #pragma unroll
    for (int kk = 0; kk < 32; kk += 4) {
      v2f a = *(const v2f*)&XiT[w * 16 + m][kk + kb];   // A[m][k]=X[d0+k][i0+m]
      #pragma unroll
      for (int nt = 0; nt < 4; ++nt) {
        v2f bb = *(const v2f*)&XjT[nt * 16 + m][kk + kb];
        acc[nt] = wmma4(a, bb, acc[nt]);
      }
    }
  }
  float* gb = G + (size_t)b * C_ * C_;
  #pragma unroll
  for (int nt = 0; nt < 4; ++nt)
    #pragma unroll
    for (int v = 0; v < 8; ++v) {
      int gi = ti * 64 + w * 16 + v + hl * 8;
      int gj = tj * 64 + nt * 16 + m;
      gb[(size_t)gi * C_ + gj] = acc[nt][v];
    }
}

// ---------------------------------------------------------------------------
// K2: one workgroup (256 thr, 8 waves) per (b,h):
//   GK = G_b Wk_h -> ws; S = scale*Wq_h^T GK -> LDS; softmax; P = A^T Wout_h.
// ---------------------------------------------------------------------------
__global__ __launch_bounds__(256) void head_kernel(const float* __restrict__ wqkv,
                                                   const float* __restrict__ wout,
                                                   const float* __restrict__ G,
                                                   float* __restrict__ GK,
                                                   float* __restrict__ P) {
  __shared__ float S[64][64];
  const int h = blockIdx.x, b = blockIdx.y;
  const int t = threadIdx.x, lane = t & 31, w = t >> 5;
  const int m = lane & 15, hl = lane >> 4, kb = hl * 2;
  const float* gb = G + (size_t)b * C_ * C_;
  float* gk = GK + (size_t)(b * NH_ + h) * C_ * DH_;

  { // Step 1: GK[256,64] = G_b @ Wk_h, K=256. Wave w -> rows [w*32, w*32+32)
    v8f acc[2][4] = {};
    const int mbase = w * 32;
    for (int kk = 0; kk < C_; kk += 4) {
      v2f bf[4];
      #pragma unroll
      for (int nt = 0; nt < 4; ++nt) {
        const float* wk = wqkv + (size_t)(kk + kb) * W3_ + HID_ + h * DH_ + nt * 16 + m;
        bf[nt].x = wk[0];
        bf[nt].y = wk[W3_];
      }
      #pragma unroll
      for (int mt = 0; mt < 2; ++mt) {
        int row = mbase + mt * 16 + m;
        v2f a = *(const v2f*)&gb[(size_t)row * C_ + kk + kb];
        #pragma unroll
        for (int nt = 0; nt < 4; ++nt) acc[mt][nt] = wmma4(a, bf[nt], acc[mt][nt]);
      }
    }
    #pragma unroll
    for (int mt = 0; mt < 2; ++mt)
      for (int nt = 0; nt < 4; ++nt)
        for (int v = 0; v < 8; ++v)
          gk[(size_t)(mbase + mt * 16 + v + hl * 8) * DH_ + nt * 16 + m] = acc[mt][nt][v];
  }
  __syncthreads();

  { // Step 2: S[64,64] = scale * Wq_h^T @ GK, K=256. wave -> (mt, 2 n-tiles)
    const int mt = w >> 1, ntb = (w & 1) * 2;
    v8f acc[2] = {};
    for (int kk = 0; kk < C_; kk += 4) {
      v2f a;                                  // A[i][c] = Wq[c][h*64+i]
      const float* wq = wqkv + (size_t)(kk + kb) * W3_ + h * DH_ + mt * 16 + m;
      a.x = wq[0];
      a.y = wq[W3_];
      #pragma unroll
      for (int u = 0; u < 2; ++u) {
        v2f bb;
        const float* g2 = gk + (size_t)(kk + kb) * DH_ + (ntb + u) * 16 + m;
        bb.x = g2[0];
        bb.y = g2[DH_];
        acc[u] = wmma4(a, bb, acc[u]);
      }
    }
    #pragma unroll
    for (int u = 0; u < 2; ++u)
      for (int v = 0; v < 8; ++v)
        S[mt * 16 + v + hl * 8][(ntb + u) * 16 + m] = acc[u][v] * SCALE_;
  }
  __syncthreads();

  // Step 3: row softmax (axis j), one thread per row
  if (t < 64) {
    float mx = -1e30f;
    for (int j = 0; j < 64; ++j) mx = fmaxf(mx, S[t][j]);
    float sum = 0.f;
    for (int j = 0; j < 64; ++j) { float e = __expf(S[t][j] - mx); S[t][j] = e; sum += e; }
    float r = 1.f / sum;
    for (int j = 0; j < 64; ++j) S[t][j] *= r;
  }
  __syncthreads();

  { // Step 4: P[64,256] = A^T @ Wout_h, K=64. wave -> (j-tile, 8 n-tiles)
    const int mt = w >> 1, nh = w & 1;
    v8f acc[8] = {};
    for (int kk = 0; kk < DH_; kk += 4) {
      v2f a;                                  // A[j][i] = attn[i][j] = S[i][j]
      a.x = S[kk + kb][mt * 16 + m];
      a.y = S[kk + kb + 1][mt * 16 + m];
      #pragma unroll
      for (int u = 0; u < 8; ++u) {
        int nt = nh * 8 + u;
        v2f bb;                               // B[i][c'] = wout[h*64+i][c']
        const float* wo = wout + (size_t)(h * DH_ + kk + kb) * C_ + nt * 16 + m;
        bb.x = wo[0];
        bb.y = wo[C_];
        acc[u] = wmma4(a, bb, acc[u]);
      }
    }
    float* pb = P + (size_t)b * HID_ * C_;
    #pragma unroll
    for (int u = 0; u < 8; ++u)
      for (int v = 0; v < 8; ++v) {
        int jr = mt * 16 + v + hl * 8;
        int nc = (nh * 8 + u) * 16 + m;
        pb[(size_t)(h * DH_ + jr) * C_ + nc] = acc[u][v];
      }
  }
}

// ---------------------------------------------------------------------------
// K3: M_b[256,256] = Wv[256,512] @ P_b[512,256]. Block = 256 thr, 256x64 strip.
// ---------------------------------------------------------------------------
__global__ __launch_bounds__(256) void mmat_kernel(const float* __restrict__ wqkv,
                                                   const float* __restrict__ P,
                                                   float* __restrict__ M) {
  const int n0 = blockIdx.x * 64, b = blockIdx.y;
  const int t = threadIdx.x, lane = t & 31, w = t >> 5;
  const int m = lane & 15, hl = lane >> 4, kb = hl * 2;
  const float* pb = P + (size_t)b * HID_ * C_;
  v8f acc[2][4] = {};
  const int mbase = w * 32;
  for (int kk = 0; kk < HID_; kk += 4) {
    v2f bf[4];
    #pragma unroll
    for (int nt = 0; nt < 4; ++nt) {
      const float* pp = pb + (size_t)(kk + kb) * C_ + n0 + nt * 16 + m;
      bf[nt].x = pp[0];
      bf[nt].y = pp[C_];
    }
    #pragma unroll
    for (int mt = 0; mt < 2; ++mt) {
      int row = mbase + mt * 16 + m;
      v2f a = *(const v2f*)&wqkv[(size_t)row * W3_ + 1024 + kk + kb];  // Wv[c][q]
      #pragma unroll
      for (int nt = 0; nt < 4; ++nt) acc[mt][nt] = wmma4(a, bf[nt], acc[mt][nt]);
    }
  }
  float* mb = M + (size_t)b * C_ * C_;
  #pragma unroll
  for (int mt = 0; mt < 2; ++mt)
    for (int nt = 0; nt < 4; ++nt)
      for (int v = 0; v < 8; ++v)
        mb[(size_t)(mbase + mt * 16 + v + hl * 8) * C_ + n0 + nt * 16 + m] = acc[mt][nt][v];
}

// ---------------------------------------------------------------------------
// K4: y_b = X_b @ M_b + b_out. Block = 128 thr (4 waves) -> 64x64 output tile.
// Double-buffered K-panels (16 wide) streamed via async global->LDS b128.
// X panel kept [m][k] (pitch 20): contiguous for b128 staging AND gives each
// A-fragment as one aligned b64 LDS load.
// ---------------------------------------------------------------------------
__global__ __launch_bounds__(128) void out_kernel(const float* __restrict__ x,
                                                  const float* __restrict__ M,
                                                  const float* __restrict__ bout,
                                                  float* __restrict__ y) {
  __shared__ float Xs[2][64][20];  // [buf][m][k]
  __shared__ float Ms[2][16][68];  // [buf][k][n]
  const int m0 = blockIdx.x * 64, n0 = blockIdx.y * 64, b = blockIdx.z;
  const int t = threadIdx.x, lane = t & 31, w = t >> 5;
  const int m = lane & 15, hl = lane >> 4, kb = hl * 2;
  const float* xb = x + (size_t)b * HW_ * C_;
  const float* mb = M + (size_t)b * C_ * C_;

  // stage one K-panel (k0) into buffer `buf`
  auto stage = [&](int k0, int buf) {
    #pragma unroll
    for (int s = 0; s < 2; ++s) {            // X: 64 rows x 16 k = 256 x 16B
      int f = t + s * 128;
      int r = f >> 2, c4 = (f & 3) * 4;
      cp16(&xb[(size_t)(m0 + r) * C_ + k0 + c4], &Xs[buf][r][c4]);
    }
    #pragma unroll
    for (int s = 0; s < 2; ++s) {            // M: 16 rows x 64 n = 256 x 16B
      int f = t + s * 128;
      int r = f >> 4, c4 = (f & 15) * 4;
      cp16(&mb[(size_t)(k0 + r) * C_ + n0 + c4], &Ms[buf][r][c4]);
    }
  };

  v8f acc[4] = {};
  stage(0, 0);
  wait_cp();
  __syncthreads();

  constexpr int NCHUNK = C_ / 16;            // 16
  for (int c = 0; c < NCHUNK; ++c) {
    const int buf = c & 1;
    if (c + 1 < NCHUNK) stage((c + 1) * 16, buf ^ 1);   // overlap next panel
    #pragma unroll
    for (int kk = 0; kk < 16; kk += 4) {
      v2f a = *(const v2f*)&Xs[buf][w * 16 + m][kk + kb];
      #pragma unroll
      for (int nt = 0; nt < 4; ++nt) {
        v2f bb;
        bb.x = Ms[buf][kk + kb][nt * 16 + m];
        bb.y = Ms[buf][kk + kb + 1][nt * 16 + m];
        acc[nt] = wmma4(a, bb, acc[nt]);
      }
    }
    if (c + 1 < NCHUNK) {
      wait_cp();
      __syncthreads();
    }
  }
  #pragma unroll
  for (int nt = 0; nt < 4; ++nt)
    for (int v = 0; v < 8; ++v) {
      int gm = m0 + w * 16 + v + hl * 8;
      int gn = n0 + nt * 16 + m;
      y[((size_t)b * HW_ + gm) * C_ + gn] = acc[nt][v] + bout[gn];
    }
}

extern "C" void kernel_launch(void* const* d_in, const int* in_sizes, int n_in,
                              void* d_out, int out_size, void* d_ws, size_t ws_size,
                              hipStream_t stream) {
  const float* x    = (const float*)d_in[0];
  const float* wqkv = (const float*)d_in[1];
  const float* wout = (const float*)d_in[2];
  const float* bout = (const float*)d_in[3];
  float* y  = (float*)d_out;
  float* ws = (float*)d_ws;
  float* G  = ws;              // 16*256*256      = 1,048,576 f
  float* GK = G  + 1048576;    // 16*8*256*64     = 2,097,152 f
  float* P  = GK + 2097152;    // 16*512*256      = 2,097,152 f
  float* M  = P  + 2097152;    // 16*256*256      = 1,048,576 f  (24 MiB total)

  gram_kernel<<<dim3(4, 4, 16), 128, 0, stream>>>(x, G);
  head_kernel<<<dim3(8, 16),    256, 0, stream>>>(wqkv, wout, G, GK, P);
  mmat_kernel<<<dim3(4, 16),    256, 0, stream>>>(wqkv, P, M);
  out_kernel <<<dim3(64, 4, 16),128, 0, stream>>>(x, M, bout, y);
}